// CausalSelfAttention_50276887167444
// MI455X (gfx1250) — compile-verified
//
#include <hip/hip_runtime.h>
#include <hip/hip_bf16.h>
#include <math.h>

typedef __attribute__((ext_vector_type(2))) float v2f;
typedef __attribute__((ext_vector_type(8))) float v8f;

// fp32 WMMA: D(16x16) = A(16x4) * B(4x16) + C   (full fp32 precision path)
__device__ __forceinline__ v8f wmma_f32x4(v2f a, v2f b, v8f c) {
  return __builtin_amdgcn_wmma_f32_16x16x4_f32(false, a, false, b, (short)0, c,
                                               false, false);
}

// Wave-local LDS fence (write->read visibility within one wave).
#define WAVE_LDS_FENCE() asm volatile("s_wait_dscnt 0" ::: "memory")

// CDNA5 async global->LDS copy (ASYNCcnt-tracked DMA-style copy).
// LDS destination byte offset in lds_off; 16 bytes per active lane.
__device__ __forceinline__ void async_load_b128(unsigned lds_off,
                                                const float* gaddr) {
  asm volatile("global_load_async_to_lds_b128 %0, %1, off"
               :: "v"(lds_off), "v"(gaddr)
               : "memory");
}
#define WAIT_ASYNC_LE4() asm volatile("s_wait_asynccnt 0x4" ::: "memory")
#define WAIT_ASYNC_0()   asm volatile("s_wait_asynccnt 0x0" ::: "memory")

// LDS byte offset of a __shared__ object (flat LDS aperture: low 32 bits).
__device__ __forceinline__ unsigned lds_off_of(const void* p) {
  return (unsigned)(uintptr_t)p;
}

// ---------------------------------------------------------------------------
// Block-cooperative GEMM + bias: C[M,N] = A[M,K] @ B[K,N] + bias[N]
// Block (256 thr / 8 waves) -> 128x128 tile; K-chunks of 16 double-buffered
// in LDS via async b128 copies. Wave (4x2 grid) -> 32x64 subtile.
// Requires M%128==0, N%128==0, K%16==0.
// ---------------------------------------------------------------------------
#define A_STRIDE 20  // 16 + pad: keeps b128 stores 16B-aligned, kills bank conflicts

__global__ __launch_bounds__(256) void gemm_bias_128x128(
    const float* __restrict__ A, const float* __restrict__ B,
    const float* __restrict__ bias, float* __restrict__ C,
    int M, int N, int K) {
  __shared__ float Abuf[2][128 * A_STRIDE];
  __shared__ float Bbuf[2][16 * 128];

  const int nblk_n = N >> 7;
  const int m0 = (blockIdx.x / nblk_n) << 7;
  const int n0 = (blockIdx.x % nblk_n) << 7;

  const int tid  = threadIdx.x;
  const int wib  = tid >> 5;          // wave in block: 0..7
  const int lane = tid & 31;
  const int half = lane >> 4;
  const int lidx = lane & 15;
  const int m_off = (wib & 3) << 5;   // wave rows: 0/32/64/96
  const int n_off = (wib >> 2) << 6;  // wave cols: 0/64

  // ---- per-thread async-copy addressing (A: 128x16 chunk, B: 16x128) ----
  const int tA_row = tid >> 2;              // 0..63 (and +64)
  const int tA_c4  = (tid & 3) << 2;        // 0,4,8,12
  const int tB_row = tid >> 5;              // 0..7 (and +8)
  const int tB_c4  = (tid & 31) << 2;       // 0..124

  const float* gA0 = A + (size_t)(m0 + tA_row) * K + tA_c4;
  const float* gA1 = A + (size_t)(m0 + tA_row + 64) * K + tA_c4;
  const float* gB0 = B + (size_t)tB_row * N + n0 + tB_c4;
  const float* gB1 = B + (size_t)(tB_row + 8) * N + n0 + tB_c4;

  const unsigned lA0 = lds_off_of(&Abuf[0][tA_row * A_STRIDE + tA_c4]);
  const unsigned lA1 = lds_off_of(&Abuf[0][(tA_row + 64) * A_STRIDE + tA_c4]);
  const unsigned lB0 = lds_off_of(&Bbuf[0][tB_row * 128 + tB_c4]);
  const unsigned lB1 = lds_off_of(&Bbuf[0][(tB_row + 8) * 128 + tB_c4]);
  const unsigned Asz = sizeof(Abuf[0]);
  const unsigned Bsz = sizeof(Bbuf[0]);

  const int nk = K >> 4;
  const size_t bstep = (size_t)N * 16;

  // prefetch chunk 0 into buffer 0
  async_load_b128(lA0, gA0);
  async_load_b128(lA1, gA1);
  async_load_b128(lB0, gB0);
  async_load_b128(lB1, gB1);

  v8f acc0[4] = {};
  v8f acc1[4] = {};

  for (int kc = 0; kc < nk; ++kc) {
    const int buf = kc & 1;
    if (kc + 1 < nk) {
      const int nb = (kc + 1) & 1;
      const int gk = (kc + 1) << 4;
      async_load_b128(lA0 + nb * Asz, gA0 + gk);
      async_load_b128(lA1 + nb * Asz, gA1 + gk);
      async_load_b128(lB0 + nb * Bsz, gB0 + (size_t)(kc + 1) * bstep);
      async_load_b128(lB1 + nb * Bsz, gB1 + (size_t)(kc + 1) * bstep);
      WAIT_ASYNC_LE4();  // chunk kc's 4 copies done; kc+1 still in flight
    } else {
      WAIT_ASYNC_0();
    }
    __syncthreads();  // all waves' chunk-kc data in LDS

#pragma unroll
    for (int ks = 0; ks < 4; ++ks) {  // 4 k-steps of 4 within the chunk
      const float* Ab =
          &Abuf[buf][(m_off + lidx) * A_STRIDE + ks * 4 + half * 2];
      const v2f a0 = *(const v2f*)Ab;
      const v2f a1 = *(const v2f*)(Ab + 16 * A_STRIDE);
      const float* Bb = &Bbuf[buf][(ks * 4 + half * 2) * 128 + n_off + lidx];
#pragma unroll
      for (int t = 0; t < 4; ++t) {
        v2f bb;
        bb.x = Bb[t * 16];
        bb.y = Bb[t * 16 + 128];
        acc0[t] = wmma_f32x4(a0, bb, acc0[t]);
        acc1[t] = wmma_f32x4(a1, bb, acc1[t]);
      }
    }
    __syncthreads();  // safe for next iteration's prefetch to overwrite this buffer
  }

#pragma unroll
  for (int t = 0; t < 4; ++t) {
    const int col = n0 + n_off + t * 16 + lidx;
    const float bv = bias[col];
#pragma unroll
    for (int v = 0; v < 8; ++v) {
      const int row = m0 + m_off + v + half * 8;
      C[(size_t)row * N + col] = acc0[t][v] + bv;
      C[(size_t)(row + 16) * N + col] = acc1[t][v] + bv;
    }
  }
}

// ---------------------------------------------------------------------------
// Sliding-window causal attention (flash-style), fp32 WMMA.
// qkv: [B, T, 3C] (Q | K | V per row). y: [B, T, C] with head-major cols.
// One wave -> one 16-query tile of one (b, h). All fragment loads are
// batched ahead of the WMMA chains so one wait covers 16 back-to-back WMMAs.
// ---------------------------------------------------------------------------
__global__ __launch_bounds__(256) void attn_window16(
    const float* __restrict__ qkv, float* __restrict__ y) {
  constexpr int T = 2048, Cc = 768, H = 12, HD = 64, WIN = 256;
  constexpr int STRIDE = 3 * Cc;
  constexpr float SCALE = 0.125f;  // 1/sqrt(64)
  constexpr float NEG = -3.0e38f;

  const int wib  = threadIdx.x >> 5;
  const int wave = blockIdx.x * (blockDim.x >> 5) + wib;
  const int lane = threadIdx.x & 31;
  const int half = lane >> 4;
  const int lidx = lane & 15;

  const int qt = wave & 127;  // T/16 = 128 query tiles
  const int bh = wave >> 7;
  const int h  = bh % H;
  const int b  = bh / H;
  const int q0 = qt << 4;

  __shared__ float lds[8 * 320];
  float* Sp   = lds + wib * 320;
  float* rowm = Sp + 256;
  float* rowl = Sp + 272;
  float* cfv  = Sp + 288;

  if (lane < 16) { rowm[lane] = NEG; rowl[lane] = 0.0f; }
  WAVE_LDS_FENCE();

  // Preload Q fragments for all 16 k-steps (hd=64 / 4).
  const float* Qrow = qkv + (size_t)(b * T + q0 + lidx) * STRIDE + h * HD;
  v2f qa[16];
#pragma unroll
  for (int kk = 0; kk < 16; ++kk) {
    qa[kk].x = Qrow[kk * 4 + half * 2];
    qa[kk].y = Qrow[kk * 4 + half * 2 + 1];
  }

  v8f o[4] = {};

  const int kmin  = (q0 - (WIN - 1)) > 0 ? (q0 - (WIN - 1)) : 0;
  const int kt_lo = kmin >> 4;
  const int kt_hi = q0 >> 4;

  for (int kt = kt_lo; kt <= kt_hi; ++kt) {
    const int k0 = kt << 4;

    // ---- S = Q @ K^T : batch 16 B-fragment loads, then 16 WMMAs ----
    const float* Krow =
        qkv + (size_t)(b * T + k0 + lidx) * STRIDE + Cc + h * HD;
    v2f kb[16];
#pragma unroll
    for (int kk = 0; kk < 16; ++kk) {
      kb[kk].x = Krow[kk * 4 + half * 2];
      kb[kk].y = Krow[kk * 4 + half * 2 + 1];
    }
    v8f s = {};
#pragma unroll
    for (int kk = 0; kk < 16; ++kk) s = wmma_f32x4(qa[kk], kb[kk], s);

    // ---- scale + sliding-window causal mask, stage into LDS ----
    const int kcol = k0 + lidx;
#pragma unroll
    for (int v = 0; v < 8; ++v) {
      const int qrow = q0 + v + half * 8;
      const bool keep = (kcol <= qrow) && (kcol + WIN > qrow);
      Sp[(v + half * 8) * 16 + lidx] = keep ? s[v] * SCALE : NEG;
    }
    WAVE_LDS_FENCE();

    // ---- online softmax: lane r (<16) owns query row r ----
    if (lane < 16) {
      float mx = NEG;
#pragma unroll
      for (int j = 0; j < 16; ++j) mx = fmaxf(mx, Sp[lane * 16 + j]);
      const float mold = rowm[lane];
      const float mnew = fmaxf(mold, mx);
      const float cf = expf(mold - mnew);
      float sum = 0.0f;
#pragma unroll
      for (int j = 0; j < 16; ++j) {
        const float sv = Sp[lane * 16 + j];
        float p = expf(sv - mnew);
        p = (sv < -1.0e37f) ? 0.0f : p;  // fully-masked entries -> 0
        Sp[lane * 16 + j] = p;
        sum += p;
      }
      rowl[lane] = rowl[lane] * cf + sum;
      rowm[lane] = mnew;
      cfv[lane] = cf;
    }
    WAVE_LDS_FENCE();

    // ---- rescale O by cf[row] ----
#pragma unroll
    for (int v = 0; v < 8; ++v) {
      const float cf = cfv[v + half * 8];
#pragma unroll
      for (int t = 0; t < 4; ++t) o[t][v] *= cf;
    }

    // ---- O += P @ V : batch P fragments + 16 V fragments, then 16 WMMAs ----
    v2f pa[4];
#pragma unroll
    for (int kk = 0; kk < 4; ++kk) {
      pa[kk].x = Sp[lidx * 16 + kk * 4 + half * 2];
      pa[kk].y = Sp[lidx * 16 + kk * 4 + half * 2 + 1];
    }
    v2f vb[16];
#pragma unroll
    for (int kk = 0; kk < 4; ++kk) {
      const float* V0 = qkv +
          (size_t)(b * T + k0 + kk * 4 + half * 2) * STRIDE + 2 * Cc +
          h * HD + lidx;
#pragma unroll
      for (int t = 0; t < 4; ++t) {
        vb[kk * 4 + t].x = V0[t * 16];
        vb[kk * 4 + t].y = V0[t * 16 + STRIDE];
      }
    }
#pragma unroll
    for (int kk = 0; kk < 4; ++kk)
#pragma unroll
      for (int t = 0; t < 4; ++t)
        o[t] = wmma_f32x4(pa[kk], vb[kk * 4 + t], o[t]);

    WAVE_LDS_FENCE();  // Sp reused next iteration
  }

  // ---- normalize by running sum and store y[b, q, h*64 + d] ----
#pragma unroll
  for (int v = 0; v < 8; ++v) {
    const float inv = 1.0f / rowl[v + half * 8];
    const int qrow = q0 + v + half * 8;
    float* yr = y + (size_t)(b * T + qrow) * Cc + h * HD + lidx;
#pragma unroll
    for (int t = 0; t < 4; ++t) yr[t * 16] = o[t][v] * inv;
  }
}

// ---------------------------------------------------------------------------
extern "C" void kernel_launch(void* const* d_in, const int* in_sizes, int n_in,
                              void* d_out, int out_size, void* d_ws, size_t ws_size,
                              hipStream_t stream) {
  const float* x      = (const float*)d_in[0];  // [4,2048,768]
  const float* W_attn = (const float*)d_in[1];  // [768,2304]
  const float* b_attn = (const float*)d_in[2];  // [2304]
  const float* W_proj = (const float*)d_in[3];  // [768,768]
  const float* b_proj = (const float*)d_in[4];  // [768]
  float* out = (float*)d_out;                   // [4,2048,768]

  constexpr int Bz = 4, T = 2048, C = 768;
  constexpr int M = Bz * T;                     // 8192 rows
  float* qkv  = (float*)d_ws;                   // [M, 2304]
  float* ybuf = qkv + (size_t)M * (3 * C);      // [M, 768]

  // 1) qkv = x @ W_attn + b_attn   (128x128 tiles: 64 x 18 blocks)
  {
    const int N = 3 * C, K = C;
    const int blocks = (M / 128) * (N / 128);
    gemm_bias_128x128<<<blocks, 256, 0, stream>>>(x, W_attn, b_attn, qkv, M, N, K);
  }
  // 2) windowed attention -> ybuf [B,T,C]
  {
    const int waves = Bz * 12 * (T / 16);  // 6144
    attn_window16<<<waves / 8, 256, 0, stream>>>(qkv, ybuf);
  }
  // 3) out = ybuf @ W_proj + b_proj   (64 x 6 blocks)
  {
    const int N = C, K = C;
    const int blocks = (M / 128) * (N / 128);
    gemm_bias_128x128<<<blocks, 256, 0, stream>>>(ybuf, W_proj, b_proj, out, M, N, K);
  }
}